// Net2_57939108823553
// MI455X (gfx1250) — compile-verified
//
#include <hip/hip_runtime.h>

#define N 1024
#define HID 64
#define CCH 32          // channel chunk staged in LDS at a time

typedef __attribute__((ext_vector_type(2))) float v2f;
typedef __attribute__((ext_vector_type(8))) float v8f;

// ---------------------------------------------------------------------------
// Fused CNN: y = conv2(relu(conv1(x) + b1)) + b2   (both 3x3, SAME zero-pad)
// One 16x16 output tile per 256-thread block. Input patch (20x20, halo 2) and
// the ReLU'd hidden activations (18x18 x CCH channels) are staged in LDS so
// the 64-channel hidden tensor (256 MB in fp32!) never touches HBM.
// ---------------------------------------------------------------------------
__global__ __launch_bounds__(256) void cnn_fused_kernel(
    const float* __restrict__ x,
    const float* __restrict__ W1, const float* __restrict__ b1,
    const float* __restrict__ W2, const float* __restrict__ b2,
    float* __restrict__ y) {
  __shared__ float sX[20][20];            // input patch with halo 2
  __shared__ float sH[18 * 18][CCH];      // hidden (relu(conv1)) chunk: ~41.5 KB
  __shared__ float sW1[HID * 9];
  __shared__ float sW2[HID * 9];
  __shared__ float sB1[HID];

  const int tid = threadIdx.x;
  const int tx = tid & 15;
  const int ty = tid >> 4;
  const int bx = blockIdx.x & 63;         // N/16 = 64 tiles per row
  const int by = blockIdx.x >> 6;
  const int ox0 = bx * 16;
  const int oy0 = by * 16;

  // weights -> LDS
  for (int i = tid; i < HID * 9; i += 256) { sW1[i] = W1[i]; sW2[i] = W2[i]; }
  if (tid < HID) sB1[tid] = b1[tid];

  // input patch -> LDS (zero pad = SAME)
  for (int i = tid; i < 20 * 20; i += 256) {
    const int py = i / 20, px = i % 20;
    const int gy = oy0 + py - 2, gx = ox0 + px - 2;
    float v = 0.0f;
    if (gy >= 0 && gy < N && gx >= 0 && gx < N) v = x[gy * N + gx];
    sX[py][px] = v;
  }
  __syncthreads();

  float acc = b2[0];

  for (int c0 = 0; c0 < HID; c0 += CCH) {
    // stage 1: hidden activations for 18x18 halo'd pixels, channels [c0, c0+CCH)
    for (int p = tid; p < 18 * 18; p += 256) {
      const int py = p / 18, px = p % 18;   // hidden pixel = global (oy0+py-1, ox0+px-1)
      float in[9];
#pragma unroll
      for (int t = 0; t < 9; ++t) in[t] = sX[py + t / 3][px + t % 3];
#pragma unroll 4
      for (int c = 0; c < CCH; ++c) {
        float h = sB1[c0 + c];
#pragma unroll
        for (int t = 0; t < 9; ++t) h = fmaf(sW1[(c0 + c) * 9 + t], in[t], h);
        sH[p][c] = fmaxf(h, 0.0f);
      }
    }
    __syncthreads();

    // stage 2: conv2 (HID->1) partial accumulation over this channel chunk
#pragma unroll 4
    for (int c = 0; c < CCH; ++c) {
#pragma unroll
      for (int t = 0; t < 9; ++t) {
        acc = fmaf(sW2[(c0 + c) * 9 + t],
                   sH[(ty + t / 3) * 18 + (tx + t % 3)][c], acc);
      }
    }
    __syncthreads();  // protect sH before next chunk overwrites it
  }

  y[(oy0 + ty) * N + (ox0 + tx)] = acc;
}

// ---------------------------------------------------------------------------
// C = A @ B  (1024x1024x1024, fp32) via V_WMMA_F32_16X16X4_F32.
// 2x2 register blocking: one wave32 computes a 32x32 C macro-tile with four
// independent accumulator chains (acc00/01/10/11), reusing each A/B fragment
// twice per K-step. Doubles FLOP-per-load vs the naive 1-tile/wave version
// and gives the matrix pipe 4-way WMMA ILP inside a single wave.
// VGPR layouts per ISA 7.12.2 (32-bit A 16x4 / B 4x16 / C 16x16).
// ---------------------------------------------------------------------------
__global__ __launch_bounds__(256) void mm_wmma_f32_kernel(
    const float* __restrict__ A,   // mat
    const float* __restrict__ B,   // h
    float* __restrict__ C) {
  const int lane = threadIdx.x & 31;
  const int wave = threadIdx.x >> 5;                 // 0..7
  const int task = blockIdx.x * 8 + wave;            // 0..1023
  const int tm = (task >> 5) * 32;                   // 32 macro-tiles per dim
  const int tn = (task & 31) * 32;

  const int half = lane >> 4;                        // 0: K pair {0,1}, 1: {2,3}
  const int l = lane & 15;

  // A fragments: lane (half,l) supplies A[row][k0 + 2*half + {0,1}] (contiguous)
  const float* arow0 = A + (size_t)(tm + l) * N + 2 * half;
  const float* arow1 = arow0 + (size_t)16 * N;
  // B fragments: lane (half,l) supplies B[k0 + 2*half + {0,1}][col] (stride-N)
  const float* bcol0 = B + (size_t)(2 * half) * N + tn + l;
  const float* bcol1 = bcol0 + 16;

  v8f acc00 = {}, acc01 = {}, acc10 = {}, acc11 = {};
  for (int k0 = 0; k0 < N; k0 += 4) {
    v2f a0, a1, b0, b1;
    a0.x = arow0[k0];
    a0.y = arow0[k0 + 1];
    a1.x = arow1[k0];
    a1.y = arow1[k0 + 1];
    const size_t koff = (size_t)k0 * N;
    b0.x = bcol0[koff];
    b0.y = bcol0[koff + N];
    b1.x = bcol1[koff];
    b1.y = bcol1[koff + N];
    // 8 args: (neg_a, A, neg_b, B, c_mod, C, reuse_a, reuse_b)
    acc00 = __builtin_amdgcn_wmma_f32_16x16x4_f32(false, a0, false, b0, (short)0, acc00, false, false);
    acc01 = __builtin_amdgcn_wmma_f32_16x16x4_f32(false, a0, false, b1, (short)0, acc01, false, false);
    acc10 = __builtin_amdgcn_wmma_f32_16x16x4_f32(false, a1, false, b0, (short)0, acc10, false, false);
    acc11 = __builtin_amdgcn_wmma_f32_16x16x4_f32(false, a1, false, b1, (short)0, acc11, false, false);
  }

  // D layout: VGPR r -> row (tileM + r + 8*half), col (tileN + l)
  float* c00 = C + (size_t)(tm + 8 * half) * N + tn + l;         // rows tm..,  cols tn..
  float* c10 = c00 + (size_t)16 * N;                             // rows tm+16, cols tn..
#pragma unroll
  for (int r = 0; r < 8; ++r) {
    c00[(size_t)r * N]      = acc00[r];
    c00[(size_t)r * N + 16] = acc01[r];
    c10[(size_t)r * N]      = acc10[r];
    c10[(size_t)r * N + 16] = acc11[r];
  }
}

// ---------------------------------------------------------------------------
// Net2.forward: 3 x ( fused-cnn -> mat @ h ), ping-pong through d_ws.
// ---------------------------------------------------------------------------
extern "C" void kernel_launch(void* const* d_in, const int* in_sizes, int n_in,
                              void* d_out, int out_size, void* d_ws, size_t ws_size,
                              hipStream_t stream) {
  const float* x   = (const float*)d_in[0];
  const float* mat = (const float*)d_in[1];
  const float* W1  = (const float*)d_in[2];
  const float* b1  = (const float*)d_in[3];
  const float* W2  = (const float*)d_in[4];
  const float* b2  = (const float*)d_in[5];
  float* out  = (float*)d_out;
  float* bufA = (float*)d_ws;            // N*N floats (4 MB)
  float* bufB = bufA + (size_t)N * N;    // N*N floats (4 MB)

  const dim3 cnnGrid(64 * 64), cnnBlk(256);
  const dim3 mmGrid(1024 / 8), mmBlk(256);   // 1024 macro-tiles, 8 waves/block

  // iter 1
  cnn_fused_kernel<<<cnnGrid, cnnBlk, 0, stream>>>(x, W1, b1, W2, b2, bufA);
  mm_wmma_f32_kernel<<<mmGrid, mmBlk, 0, stream>>>(mat, bufA, bufB);
  // iter 2
  cnn_fused_kernel<<<cnnGrid, cnnBlk, 0, stream>>>(bufB, W1, b1, W2, b2, bufA);
  mm_wmma_f32_kernel<<<mmGrid, mmBlk, 0, stream>>>(mat, bufA, bufB);
  // iter 3
  cnn_fused_kernel<<<cnnGrid, cnnBlk, 0, stream>>>(bufB, W1, b1, W2, b2, bufA);
  mm_wmma_f32_kernel<<<mmGrid, mmBlk, 0, stream>>>(mat, bufA, out);
}